// GINConv_81398220194522
// MI455X (gfx1250) — compile-verified
//
#include <hip/hip_runtime.h>

typedef __attribute__((ext_vector_type(2))) float v2f;
typedef __attribute__((ext_vector_type(4))) float v4f;
typedef __attribute__((ext_vector_type(8))) float v8f;

#define N_NODES 100000
#define DEG     16
#define C       128
#define TILE_M  16
#define HPAD    132          // padded LDS row stride (floats): 4-bank skew per row
#define LN_EPS  1e-5f
#define NB_BATCH 4           // neighbor loads in flight per batch

__launch_bounds__(256)
__global__ void gin_fused_wmma(const float* __restrict__ x,
                               const int*   __restrict__ edge,
                               const float* __restrict__ W,
                               const float* __restrict__ bvec,
                               const float* __restrict__ epsp,
                               const float* __restrict__ gamma,
                               const float* __restrict__ beta,
                               float* __restrict__ out,
                               int nTiles) {
    __shared__ float hS[TILE_M * HPAD];      // 16 x 132 combined tile
    __shared__ float partS[TILE_M][8];       // per-wave row sums
    __shared__ float partQ[TILE_M][8];       // per-wave row sumsqs
    __shared__ float muS[TILE_M];
    __shared__ float rsS[TILE_M];

    const int tid  = threadIdx.x;
    const int lane = tid & 31;
    const int wave = tid >> 5;               // 0..7 -> output col tile
    const int l16  = lane & 15;
    const int hi   = lane >> 4;              // 0/1 half of wave

    const float self_scale = 1.0f + epsp[0];

    // ---- Per-wave B fragments: cols 16*wave + l16, K-pairs (4j + 2*hi, +1) ----
    // WMMA f32 16x16x4 B layout: lanes 0-15 hold N=0..15 K={0,1}; lanes 16-31 K={2,3}
    v2f bfrag[32];
    const int colg = (wave << 4) + l16;
    #pragma unroll
    for (int j = 0; j < 32; ++j) {
        const int k = (j << 2) + (hi << 1);
        bfrag[j].x = W[(size_t)k * C + colg];
        bfrag[j].y = W[(size_t)(k + 1) * C + colg];
    }
    const float bias_l  = bvec[colg];
    const float gamma_l = gamma[colg];
    const float beta_l  = beta[colg];

    for (int t = blockIdx.x; t < nTiles; t += gridDim.x) {
        const int row0 = t * TILE_M;

        // ---- Phase 1: coalesced gather + GIN combine; wave handles rows 2w, 2w+1
        #pragma unroll
        for (int rr = 0; rr < 2; ++rr) {
            const int r    = (wave << 1) + rr;         // 0..15 within tile
            const int rowG = row0 + r;
            // lanes 0..15 each fetch one neighbor index for this row
            int vidx = edge[(size_t)rowG * DEG + l16];

            const float* xr = x + (size_t)rowG * C + (lane << 2);
            v4f acc;
            acc.x = self_scale * xr[0];
            acc.y = self_scale * xr[1];
            acc.z = self_scale * xr[2];
            acc.w = self_scale * xr[3];

            // Branchless batched gather: NB_BATCH loads in flight, then accumulate.
            // idx is wave-uniform; padding idx==N is clamped to row 0 with weight 0.
            #pragma unroll
            for (int nb = 0; nb < DEG; nb += NB_BATCH) {
                v4f   tv[NB_BATCH];
                float wg[NB_BATCH];
                #pragma unroll
                for (int u = 0; u < NB_BATCH; ++u) {
                    const int idx   = __builtin_amdgcn_readlane(vidx, nb + u);
                    const bool ok   = idx < N_NODES;            // uniform
                    const int  ci   = ok ? idx : 0;             // s_cselect
                    wg[u]           = ok ? 1.0f : 0.0f;         // s_cselect
                    tv[u] = *(const v4f*)(x + (size_t)ci * C + (lane << 2));
                }
                #pragma unroll
                for (int u = 0; u < NB_BATCH; ++u) {
                    acc.x = fmaf(wg[u], tv[u].x, acc.x);
                    acc.y = fmaf(wg[u], tv[u].y, acc.y);
                    acc.z = fmaf(wg[u], tv[u].z, acc.z);
                    acc.w = fmaf(wg[u], tv[u].w, acc.w);
                }
            }
            *(v4f*)(&hS[r * HPAD + (lane << 2)]) = acc;
        }
        __syncthreads();

        // ---- Phase 2: D = h(16x128) x W(128x16) via 32 x V_WMMA_F32_16X16X4_F32
        // A layout: lanes 0-15 hold M=0..15 K={k,k+1}; lanes 16-31 K={k+2,k+3}
        v8f c = {};
        #pragma unroll
        for (int j = 0; j < 32; ++j) {
            const int k = (j << 2) + (hi << 1);
            v2f a = *(const v2f*)(&hS[l16 * HPAD + k]);
            c = __builtin_amdgcn_wmma_f32_16x16x4_f32(
                    /*neg_a=*/false, a, /*neg_b=*/false, bfrag[j],
                    /*c_mod=*/(short)0, c, /*reuse_a=*/false, /*reuse_b=*/false);
        }
        #pragma unroll
        for (int v = 0; v < 8; ++v) c[v] += bias_l;    // bias before LN stats

        // ---- Phase 3: LayerNorm stats (row v + 8*hi lives in lane-half 'hi')
        #pragma unroll
        for (int v = 0; v < 8; ++v) {
            float s = c[v];
            float q = c[v] * c[v];
            #pragma unroll
            for (int m = 1; m < 16; m <<= 1) {         // reduce within 16-lane half
                s += __shfl_xor(s, m, 32);
                q += __shfl_xor(q, m, 32);
            }
            if (l16 == 0) {
                const int r = v + (hi << 3);
                partS[r][wave] = s;
                partQ[r][wave] = q;
            }
        }
        __syncthreads();
        if (tid < TILE_M) {
            float s = 0.f, q = 0.f;
            #pragma unroll
            for (int wv = 0; wv < 8; ++wv) { s += partS[tid][wv]; q += partQ[tid][wv]; }
            const float mu  = s * (1.0f / C);
            const float var = q * (1.0f / C) - mu * mu;
            muS[tid] = mu;
            rsS[tid] = __frsqrt_rn(var + LN_EPS);
        }
        __syncthreads();

        // ---- Phase 4: normalize + affine + store
        #pragma unroll
        for (int v = 0; v < 8; ++v) {
            const int r = v + (hi << 3);
            const float o = (c[v] - muS[r]) * rsS[r] * gamma_l + beta_l;
            out[(size_t)(row0 + r) * C + colg] = o;
        }
        __syncthreads();   // hS/partials reused next iteration
    }
}

extern "C" void kernel_launch(void* const* d_in, const int* in_sizes, int n_in,
                              void* d_out, int out_size, void* d_ws, size_t ws_size,
                              hipStream_t stream) {
    const float* x     = (const float*)d_in[0];
    const int*   edge  = (const int*)d_in[1];
    const float* W     = (const float*)d_in[2];
    const float* bvec  = (const float*)d_in[3];
    const float* eps   = (const float*)d_in[4];
    const float* gamma = (const float*)d_in[5];
    const float* beta  = (const float*)d_in[6];
    float* out = (float*)d_out;

    const int nTiles = N_NODES / TILE_M;     // 6250, exact
    int grid = nTiles < 2048 ? nTiles : 2048;
    gin_fused_wmma<<<grid, 256, 0, stream>>>(x, edge, W, bvec, eps, gamma, beta,
                                             out, nTiles);
}